// RNNModel_7275674599521
// MI455X (gfx1250) — compile-verified
//
#include <hip/hip_runtime.h>
#include <math.h>

typedef __attribute__((ext_vector_type(2))) float v2f;
typedef __attribute__((ext_vector_type(8))) float v8f;

#define M_BATCH 128
#define T_STEPS 512
#define N_HID   1024
#define N_IN    2
#define N_OUT   2
#define ALPHA_F 0.2f
#define LDS_PITCH 68   // 64 + 4 pad: kills 16-way bank conflicts, keeps 16B alignment
#define KCHUNK  64
#define NCHUNKS (N_HID / KCHUNK)   // 16

// ---------------------------------------------------------------------------
// t = 0 init: x0 = 0, u0 = tanh(0) = 0
// ---------------------------------------------------------------------------
__global__ void init_t0(float* __restrict__ x, float* __restrict__ u) {
    int g = blockIdx.x * blockDim.x + threadIdx.x;      // 0 .. 128*1024-1
    int m = g >> 10;
    int n = g & 1023;
    size_t idx = (size_t)m * T_STEPS * N_HID + n;       // time index 0
    x[idx] = 0.0f;
    u[idx] = 0.0f;
}

// Truncate a generic pointer to the 32-bit LDS byte offset (hardware maps
// flat LDS addresses via addr[31:0], per CDNA5 aperture rules).
__device__ __forceinline__ uint32_t lds_addr32(const void* p) {
    return (uint32_t)(uintptr_t)p;
}

// ---------------------------------------------------------------------------
// One recurrent step (t in 1..T-1):
//   x_t = x_{t-1} + ALPHA*(-x_{t-1} + u_{t-1} @ W_rec^T + I_{t-1} @ W_in^T + b)
//   u_t = tanh(x_t)
// Grid: (16, 8)  -> blockIdx.x = N group of 64 cols, blockIdx.y = M tile of 16
// Block: 128 threads = 4 waves; each wave owns one 16x16 output tile.
// Double-buffered LDS staging via GLOBAL_LOAD_ASYNC_TO_LDS_B128 (ASYNCcnt).
// 4 independent accumulator chains break the WMMA D->C serial dependency
// (only ~1 wave/SIMD is resident, so intra-wave ILP is the only latency cover).
// ---------------------------------------------------------------------------
__global__ __launch_bounds__(128) void rnn_step(
    const float* __restrict__ I,
    const float* __restrict__ W_in,
    const float* __restrict__ W_rec,
    const float* __restrict__ bvec,
    float* __restrict__ x_buf,   // (M,T,N) output / state
    float* __restrict__ u_buf,   // (M,T,N) output / state
    int t)
{
    __shared__ float Alds[2][16 * LDS_PITCH];  // u_{t-1} tile: 16 (m) x 64 (k)
    __shared__ float Blds[2][64 * LDS_PITCH];  // W_rec tile: 64 (n) x 64 (k)

    const int tid  = threadIdx.x;
    const int wave = tid >> 5;
    const int lane = tid & 31;
    const int half = lane >> 4;   // 0: lanes 0-15, 1: lanes 16-31
    const int idx  = lane & 15;

    const int mBase  = blockIdx.y * 16;
    const int nBase0 = blockIdx.x * 64;
    const int nBase  = nBase0 + wave * 16;

    const size_t rowStride = (size_t)T_STEPS * N_HID;
    const size_t tPrevOff  = (size_t)(t - 1) * N_HID;

    // Issue one K-chunk of async global->LDS copies (10 b128 per thread-wave).
    auto issue_chunk = [&](int kc, int buf) {
        #pragma unroll
        for (int i = 0; i < 2; ++i) {             // A: 16 x 64 floats
            int flat = tid + i * 128;             // float4 id 0..255
            int r = flat >> 4;                    // 0..15
            int c = (flat & 15) * 4;              // 0..60
            const float* gp = &u_buf[(size_t)(mBase + r) * rowStride + tPrevOff + kc + c];
            uint32_t lp = lds_addr32(&Alds[buf][r * LDS_PITCH + c]);
            asm volatile("global_load_async_to_lds_b128 %0, %1, off"
                         :: "v"(lp), "v"(gp) : "memory");
        }
        #pragma unroll
        for (int i = 0; i < 8; ++i) {             // B: 64 x 64 floats
            int flat = tid + i * 128;             // float4 id 0..1023
            int r = flat >> 4;                    // 0..63
            int c = (flat & 15) * 4;
            const float* gp = &W_rec[(size_t)(nBase0 + r) * N_HID + kc + c];
            uint32_t lp = lds_addr32(&Blds[buf][r * LDS_PITCH + c]);
            asm volatile("global_load_async_to_lds_b128 %0, %1, off"
                         :: "v"(lp), "v"(gp) : "memory");
        }
    };

    // 4 independent accumulator chains (kb mod 4) for intra-wave WMMA ILP.
    v8f acc[4];
    #pragma unroll
    for (int q = 0; q < 4; ++q) acc[q] = (v8f){0.f,0.f,0.f,0.f,0.f,0.f,0.f,0.f};

    issue_chunk(0, 0);                             // prologue: chunk 0 -> buf 0

    for (int c = 0; c < NCHUNKS; ++c) {
        const int buf = c & 1;
        if (c + 1 < NCHUNKS) {
            issue_chunk((c + 1) * KCHUNK, buf ^ 1);   // prefetch next chunk
            // 10 newest async loads (next chunk) may be in flight; the
            // previous 10 (current chunk) are complete once cnt <= 10.
            asm volatile("s_wait_asynccnt 0xa" ::: "memory");
        } else {
            asm volatile("s_wait_asynccnt 0x0" ::: "memory");
        }
        __syncthreads();   // current buffer visible to all waves

        #pragma unroll
        for (int kb = 0; kb < 16; ++kb) {
            const int ko = kb * 4 + 2 * half;  // K pair: lanes 0-15 -> {0,1}, 16-31 -> {2,3}
            v2f a = *(const v2f*)&Alds[buf][idx * LDS_PITCH + ko];                // A[m=idx][k]
            v2f b = *(const v2f*)&Blds[buf][(wave * 16 + idx) * LDS_PITCH + ko];  // B[k][n=idx]
            acc[kb & 3] = __builtin_amdgcn_wmma_f32_16x16x4_f32(
                /*neg_a=*/false, a, /*neg_b=*/false, b,
                /*c_mod=*/(short)0, acc[kb & 3], /*reuse_a=*/false, /*reuse_b=*/false);
        }
        __syncthreads();   // done reading buf before it is refilled next+1 iter
    }

    // Merge the 4 chains (identical C/D tile layouts -> element-wise add).
    const v8f accSum = (acc[0] + acc[1]) + (acc[2] + acc[3]);

    // --- epilogue: input current, bias, leaky Euler update, tanh, store x & u
    const int   n   = nBase + idx;
    const float bn  = bvec[n];
    const float wi0 = W_in[n * N_IN + 0];
    const float wi1 = W_in[n * N_IN + 1];

    #pragma unroll
    for (int v = 0; v < 8; ++v) {
        const int m = mBase + v + 8 * half;    // C/D layout: VGPR v -> row v (+8 upper half)
        const size_t irow = ((size_t)m * T_STEPS + (size_t)(t - 1)) * N_IN;
        const float inp = I[irow + 0] * wi0 + I[irow + 1] * wi1;

        const size_t prevIdx = (size_t)m * rowStride + tPrevOff + n;
        const float xp = x_buf[prevIdx];
        const float xn = xp + ALPHA_F * (-xp + accSum[v] + inp + bn);
        const float un = tanhf(xn);

        const size_t curIdx = (size_t)m * rowStride + (size_t)t * N_HID + n;
        x_buf[curIdx] = xn;
        u_buf[curIdx] = un;
    }
}

// ---------------------------------------------------------------------------
// y = u @ W_out^T : one thread per (m,t) row, both outputs at once.
// W_out is (N_OUT, N) row-major and L2-resident.
// ---------------------------------------------------------------------------
__global__ void y_kernel(const float* __restrict__ u,
                         const float* __restrict__ W_out,
                         float* __restrict__ y)
{
    const int row = blockIdx.x * blockDim.x + threadIdx.x;  // 0 .. M*T-1
    const float4* u4 = (const float4*)(u + (size_t)row * N_HID);
    const float4* w0 = (const float4*)(W_out);
    const float4* w1 = (const float4*)(W_out + N_HID);
    float s0 = 0.f, s1 = 0.f;
    for (int i = 0; i < N_HID / 4; ++i) {
        const float4 uv = u4[i];
        const float4 a  = w0[i];
        const float4 b  = w1[i];
        s0 += uv.x * a.x + uv.y * a.y + uv.z * a.z + uv.w * a.w;
        s1 += uv.x * b.x + uv.y * b.y + uv.z * b.z + uv.w * b.w;
    }
    y[row * 2 + 0] = s0;
    y[row * 2 + 1] = s1;
}

// ---------------------------------------------------------------------------
extern "C" void kernel_launch(void* const* d_in, const int* in_sizes, int n_in,
                              void* d_out, int out_size, void* d_ws, size_t ws_size,
                              hipStream_t stream)
{
    const float* I     = (const float*)d_in[0];  // (M, T, N_in)
    const float* W_in  = (const float*)d_in[1];  // (N, N_in)
    const float* W_rec = (const float*)d_in[2];  // (N, N)
    const float* b     = (const float*)d_in[3];  // (N,)
    const float* W_out = (const float*)d_in[4];  // (N_out, N)

    float* out = (float*)d_out;
    const size_t xu = (size_t)M_BATCH * T_STEPS * N_HID;
    float* x = out;                // (M,T,N)
    float* u = out + xu;           // (M,T,N)
    float* y = out + 2 * xu;       // (M,T,N_out)

    // t = 0 state
    init_t0<<<(M_BATCH * N_HID) / 256, 256, 0, stream>>>(x, u);

    // serial recurrence: one kernel per time step (device-wide sync)
    for (int t = 1; t < T_STEPS; ++t) {
        rnn_step<<<dim3(16, 8), 128, 0, stream>>>(I, W_in, W_rec, b, x, u, t);
    }

    // readout
    y_kernel<<<(M_BATCH * T_STEPS) / 256, 256, 0, stream>>>(u, W_out, y);
}